// Backflow_50311246905906
// MI455X (gfx1250) — compile-verified
//
#include <hip/hip_runtime.h>
#include <stdint.h>

typedef __attribute__((ext_vector_type(2))) float v2f;
typedef __attribute__((ext_vector_type(8))) float v8f;

#define BATCH 16
#define NPART 512
#define HID   32
#define WAVES 4
#define JCHUNK (NPART / WAVES)      // 128 j's per wave
#define XCHUNKS ((NPART * 3) / 4)   // 384 float4 chunks of the x slab

__device__ __forceinline__ float fast_tanh(float x) {
#if __has_builtin(__builtin_amdgcn_tanhf)
  return __builtin_amdgcn_tanhf(x);                 // v_tanh_f32 on gfx1250
#elif __has_builtin(__builtin_amdgcn_tanh_f32)
  return __builtin_amdgcn_tanh_f32(x);
#else
  float e = __builtin_amdgcn_exp2f(x * 2.8853900817779268f);
  return 1.0f - 2.0f * __builtin_amdgcn_rcpf(e + 1.0f);
#endif
}

__device__ __forceinline__ void async_wait0() {
#if __has_builtin(__builtin_amdgcn_s_wait_asynccnt)
  __builtin_amdgcn_s_wait_asynccnt(0);
#else
  asm volatile("s_wait_asynccnt 0x0" ::: "memory");
#endif
}

__launch_bounds__(128, 1)
__global__ void backflow_kernel(const float* __restrict__ x,
                                const float* __restrict__ ew1,
                                const float* __restrict__ eb1,
                                const float* __restrict__ ew2,
                                const float* __restrict__ eb2,
                                const float* __restrict__ mw1,
                                const float* __restrict__ mb1,
                                const float* __restrict__ mw2,
                                const float* __restrict__ mb2,
                                float* __restrict__ out) {
  __shared__ float ldsx[NPART * 3];       // whole per-batch x slab (6 KB)
  __shared__ float red[WAVES * 16 * 4];   // per-wave partial C tiles (cols 0..3)

  const int tid  = threadIdx.x;
  const int wv   = tid >> 5;
  const int lane = tid & 31;
  const int m    = lane & 15;   // A row within tile; also B column index n
  const int half = lane >> 4;   // 0/1 -> which K pair this lane holds
  const int kb   = half * 2;    // K base within the group of 4 j's

  const int tile  = blockIdx.x;        // 0..511 : (batch, i-tile)
  const int b     = tile >> 5;
  const int ibase = (tile & 31) * 16;

  const float* __restrict__ xb = x + b * (NPART * 3);

  // ---- async-stage the 6KB x slab into LDS (ASYNCcnt path) ----
#pragma unroll
  for (int c = tid; c < XCHUNKS; c += 128) {
    const unsigned lds_addr = (unsigned)(uintptr_t)(&ldsx[c * 4]);
    const float* gp = xb + c * 4;
    asm volatile("global_load_async_to_lds_b128 %0, %1, off"
                 :: "v"(lds_addr), "v"(gp) : "memory");
  }

  // eta-MLP weights (wave-uniform scalar loads, overlap the async copy)
  float w1[HID], b1[HID], w2[HID];
#pragma unroll
  for (int k = 0; k < HID; ++k) { w1[k] = ew1[k]; b1[k] = eb1[k]; w2[k] = ew2[k]; }
  const float bias2 = eb2[0];

  // branchless one-hot selectors for the B fragment (column n = m)
  const float selx = (m == 0) ? 1.0f : 0.0f;
  const float sely = (m == 1) ? 1.0f : 0.0f;
  const float selz = (m == 2) ? 1.0f : 0.0f;
  const float selw = (m == 3) ? 1.0f : 0.0f;

  async_wait0();
  __syncthreads();

  // x_i for this lane's row (constant over the j loop) -- from LDS
  const float xi0 = ldsx[(ibase + m) * 3 + 0];
  const float xi1 = ldsx[(ibase + m) * 3 + 1];
  const float xi2 = ldsx[(ibase + m) * 3 + 2];

  v8f acc = {};   // C: cols 0..2 = sum_j s*x_j, col 3 = sum_j s (ones column)

  const int j_begin = wv * JCHUNK;
  const int j_end   = j_begin + JCHUNK;
  for (int jb = j_begin; jb < j_end; jb += 4) {
    const int j0 = jb + kb;                   // this lane's two j's: j0, j0+1
    const float* __restrict__ xj = &ldsx[j0 * 3];
    const float a0 = xj[0], a1 = xj[1], a2 = xj[2];
    const float c0 = xj[3], c1 = xj[4], c2 = xj[5];

    float dx = xi0 - a0, dy = xi1 - a1, dz = xi2 - a2;
    const float d0 = __builtin_sqrtf(__builtin_fmaf(dx, dx, __builtin_fmaf(dy, dy, dz * dz)));
    dx = xi0 - c0; dy = xi1 - c1; dz = xi2 - c2;
    const float d1 = __builtin_sqrtf(__builtin_fmaf(dx, dx, __builtin_fmaf(dy, dy, dz * dz)));

    // scalar MLP: s = w2 . tanh(d*w1 + b1) + b2   (diagonal j==i self-cancels)
    float s0 = bias2, s1 = bias2;
#pragma unroll
    for (int k = 0; k < HID; ++k) {
      s0 = __builtin_fmaf(w2[k], fast_tanh(__builtin_fmaf(d0, w1[k], b1[k])), s0);
      s1 = __builtin_fmaf(w2[k], fast_tanh(__builtin_fmaf(d1, w1[k], b1[k])), s1);
    }

    // A fragment (16x4 f32): lane holds S[m, kb], S[m, kb+1]
    v2f A; A.x = s0; A.y = s1;

    // B fragment (4x16 f32): branchless select of {x0,x1,x2,1,0...} for column m
    const float bx = __builtin_fmaf(selx, a0,
                     __builtin_fmaf(sely, a1,
                     __builtin_fmaf(selz, a2, selw)));
    const float by = __builtin_fmaf(selx, c0,
                     __builtin_fmaf(sely, c1,
                     __builtin_fmaf(selz, c2, selw)));
    v2f Bf; Bf.x = bx; Bf.y = by;

    acc = __builtin_amdgcn_wmma_f32_16x16x4_f32(
        /*neg_a=*/false, A, /*neg_b=*/false, Bf,
        /*c_mod=*/(short)0, acc, /*reuse_a=*/false, /*reuse_b=*/false);
  }

  // Spill this wave's useful C columns (0..3) to LDS
#pragma unroll
  for (int v = 0; v < 8; ++v) {
    const int row = v + 8 * half;   // C layout: lanes<16 -> M=v, lanes>=16 -> M=v+8
    if (m < 4) red[wv * 64 + row * 4 + m] = acc[v];
  }
  __syncthreads();

  // Cross-wave reduction: 64 (row,col) slots
  if (tid < 64) {
    red[tid] = red[tid] + red[64 + tid] + red[128 + tid] + red[192 + tid];
  }
  __syncthreads();

  // Epilogue: 16 rows -> outputs, plus one-body mu-MLP term
  if (tid < 16) {
    const int row = tid;
    const float s0 = red[row * 4 + 0];
    const float s1 = red[row * 4 + 1];
    const float s2 = red[row * 4 + 2];
    const float ss = red[row * 4 + 3];     // sum_j s_ij

    const float yi0 = ldsx[(ibase + row) * 3 + 0];
    const float yi1 = ldsx[(ibase + row) * 3 + 1];
    const float yi2 = ldsx[(ibase + row) * 3 + 2];

    const float di = __builtin_sqrtf(
        __builtin_fmaf(yi0, yi0, __builtin_fmaf(yi1, yi1, yi2 * yi2)));
    float mu = mb2[0];
#pragma unroll
    for (int k = 0; k < HID; ++k)
      mu = __builtin_fmaf(mw2[k], fast_tanh(__builtin_fmaf(di, mw1[k], mb1[k])), mu);

    const float g = ss + mu;   // (sum_j s)*x_i + mu*x_i
    float* __restrict__ op = out + (b * NPART + ibase + row) * 3;
    op[0] = __builtin_fmaf(g, yi0, -s0);
    op[1] = __builtin_fmaf(g, yi1, -s1);
    op[2] = __builtin_fmaf(g, yi2, -s2);
  }
}

extern "C" void kernel_launch(void* const* d_in, const int* in_sizes, int n_in,
                              void* d_out, int out_size, void* d_ws, size_t ws_size,
                              hipStream_t stream) {
  (void)in_sizes; (void)n_in; (void)d_ws; (void)ws_size; (void)out_size;
  const float* x_p = (const float*)d_in[0];
  const float* ew1 = (const float*)d_in[1];
  const float* eb1 = (const float*)d_in[2];
  const float* ew2 = (const float*)d_in[3];
  const float* eb2 = (const float*)d_in[4];
  const float* mw1 = (const float*)d_in[5];
  const float* mb1 = (const float*)d_in[6];
  const float* mw2 = (const float*)d_in[7];
  const float* mb2 = (const float*)d_in[8];
  float* out = (float*)d_out;

  const int n_tiles = (BATCH * NPART) / 16;   // 512 blocks, 4 waves each
  backflow_kernel<<<n_tiles, WAVES * 32, 0, stream>>>(
      x_p, ew1, eb1, ew2, eb2, mw1, mb1, mw2, mb2, out);
}